// HierDDmx_28355374088366
// MI455X (gfx1250) — compile-verified
//
#include <hip/hip_runtime.h>

#define BATCH 8
#define SEQ   4096
#define DIM   128
#define BASIS 64
#define LS    132   /* padded LDS row stride in floats (128 + 4) to dodge bank conflicts */
#define PI2F  6.28318530717958647692f

typedef float v2f __attribute__((ext_vector_type(2)));
typedef float v8f __attribute__((ext_vector_type(8)));

__device__ __forceinline__ float cos2pi(float ratio) {
    // cos(2*pi*ratio) with explicit range reduction so v_cos_f32 stays in range
    ratio -= floorf(ratio);
    return __builtin_amdgcn_cosf(PI2F * ratio);
}

// CDNA5 async global->LDS copy (ASYNCcnt-tracked, no VGPR data round trip).
// Generic pointers into LDS carry the workgroup-relative LDS offset in their
// low 32 bits (ISA 10.2 aperture mapping), which is exactly what the
// VDST "LDS address" operand wants (it is added to the LDS base and the instruction offset).
__device__ __forceinline__ void async_g2l_b128(void* lds_dst, const void* gsrc) {
    uint32_t loff = (uint32_t)(uintptr_t)lds_dst;
    asm volatile("global_load_async_to_lds_b128 %0, %1, off"
                 :: "v"(loff), "v"(gsrc) : "memory");
}
__device__ __forceinline__ void wait_async0() {
    asm volatile("s_wait_asynccnt 0x0" ::: "memory");
}

// ---------------------------------------------------------------------------
// layer_ab: residual = x@R.T ; normalized = LN(x@W.T) ; sc = <normalized, Bv[s]>
//           out = sc * A[:, s%64] + residual
// One block = 16 consecutive tokens x 128 outputs. 8 waves, one 16x16 N-tile each.
// ---------------------------------------------------------------------------
__global__ __launch_bounds__(256)
void k_ab(const float* __restrict__ xin,
          const float* __restrict__ W, const float* __restrict__ R,
          const float* __restrict__ A,
          const float* __restrict__ g, const float* __restrict__ b,
          float* __restrict__ out) {
    __shared__ float xs [16*LS];
    __shared__ float lin[16*LS];
    __shared__ float res[16*LS];

    const int tid  = threadIdx.x;
    const int lane = tid & 31;
    const int wv   = tid >> 5;        // wave id == output N tile (0..7)
    const int lr   = lane & 15;
    const int hh   = lane >> 4;       // half-wave: selects K pair within k-tile
    const int t0   = blockIdx.x * 16; // first flattened token of tile

    // ---- stage x tile (16 x 128) into LDS via async-to-LDS, 8 floats/thread ----
    {
        const int r  = tid >> 4;
        const int c0 = (tid & 15) * 8;
        const float* gp = xin + (size_t)(t0 + r) * DIM + c0;
        async_g2l_b128(&xs[r*LS + c0    ], gp);
        async_g2l_b128(&xs[r*LS + c0 + 4], gp + 4);
    }
    wait_async0();
    __syncthreads();

    // ---- dual fp32 GEMM: accW = x@W.T tile, accR = x@R.T tile (K = 128) ----
    v8f accW = {}; v8f accR = {};
    const float* Wp = W + (size_t)(wv*16 + lr) * DIM + 2*hh;
    const float* Rp = R + (size_t)(wv*16 + lr) * DIM + 2*hh;
    const float* Ap = &xs[lr*LS + 2*hh];
    __builtin_prefetch(Wp, 0, 0);
    __builtin_prefetch(Rp, 0, 0);
#pragma unroll 8
    for (int kt = 0; kt < 32; ++kt) {
        v2f a  = *(const v2f*)(Ap + kt*4);   // A frag: lane holds x[M=lr, K=4kt+2hh+{0,1}]
        v2f bw = *(const v2f*)(Wp + kt*4);   // B frag: B[k,n] = W[n,k]
        v2f br = *(const v2f*)(Rp + kt*4);
        accW = __builtin_amdgcn_wmma_f32_16x16x4_f32(false, a, false, bw,
                                                     (short)0, accW, false, false);
        accR = __builtin_amdgcn_wmma_f32_16x16x4_f32(false, a, false, br,
                                                     (short)0, accR, false, false);
    }
    // C/D layout: VGPR v, lane l -> D[M = v + 8*(l/16), N = l%16]
#pragma unroll
    for (int v = 0; v < 8; ++v) {
        const int m = v + 8*hh;
        lin[m*LS + wv*16 + lr] = accW[v];
        res[m*LS + wv*16 + lr] = accR[v];
    }
    __syncthreads();

    // ---- layernorm stats + basis scalar; 16 threads per row, shfl_xor reduce ----
    const int row = tid >> 4;
    const int cl  = tid & 15;
    float vals[8];
    float s1 = 0.f, s2 = 0.f;
#pragma unroll
    for (int q = 0; q < 8; ++q) {
        float vL = lin[row*LS + cl + q*16];
        vals[q] = vL; s1 += vL; s2 += vL*vL;
    }
#pragma unroll
    for (int off = 8; off; off >>= 1) {
        s1 += __shfl_xor(s1, off, 16);
        s2 += __shfl_xor(s2, off, 16);
    }
    const float mu   = s1 * (1.f/DIM);
    const float rstd = rsqrtf(s2*(1.f/DIM) - mu*mu + 1e-5f);

    const int sIdx = (t0 + row) & (SEQ - 1);
    const int j    = sIdx & (BASIS - 1);
    const float kf = (float)(j + 1);
    float sc = 0.f;
#pragma unroll
    for (int q = 0; q < 8; ++q) {
        const int cc = cl + q*16;
        float nrm = (vals[q] - mu) * rstd * g[cc] + b[cc];
        sc += nrm * cos2pi(kf / (float)(cc + 2));   // Bv[j, cc]
    }
#pragma unroll
    for (int off = 8; off; off >>= 1) sc += __shfl_xor(sc, off, 16);

    // ---- out = sc * A[:, j] + residual ----
    const int c0 = cl * 8;
    float* op = out + (size_t)(t0 + row) * DIM + c0;
    float4 o0, o1;
#pragma unroll
    for (int q = 0; q < 4; ++q)
        ((float*)&o0)[q] = sc * A[(c0 + q    ) * BASIS + j] + res[row*LS + c0 + q];
#pragma unroll
    for (int q = 0; q < 4; ++q)
        ((float*)&o1)[q] = sc * A[(c0 + 4 + q) * BASIS + j] + res[row*LS + c0 + 4 + q];
    ((float4*)op)[0] = o0;
    ((float4*)op)[1] = o1;
}

// ---------------------------------------------------------------------------
// layer_pm: lin = x@M.T ; xt = LN(lin) ; Nk[b,s,i] = sum_j xt * P[i,j]*phi[s,i,j]
//           out = Nk + lin
// One block = 2 s-positions x 8 batch rows, so the cos-heavy Peff = P .* phi[s]
// is built once per s in LDS and amortized over the whole batch (8x fewer cos).
// ---------------------------------------------------------------------------
__global__ __launch_bounds__(256)
void k_pm(const float* __restrict__ xin,
          const float* __restrict__ M, const float* __restrict__ P,
          const float* __restrict__ g, const float* __restrict__ b,
          float* __restrict__ out) {
    __shared__ float xs  [16*LS];
    __shared__ float lin [16*LS];
    __shared__ float xt  [16*LS];
    __shared__ float peff[2*16*130];   // [s of pair][i_local][j], padded stride 130

    const int tid  = threadIdx.x;
    const int lane = tid & 31;
    const int wv   = tid >> 5;
    const int lr   = lane & 15;
    const int hh   = lane >> 4;
    const int s0   = blockIdx.x * 2;

    // row m -> token (m&7)*SEQ + s0 + (m>>3); staged via async-to-LDS
    {
        const int r  = tid >> 4;
        const int c0 = (tid & 15) * 8;
        const size_t tok = (size_t)(r & 7) * SEQ + (size_t)(s0 + (r >> 3));
        const float* gp = xin + tok * DIM + c0;
        async_g2l_b128(&xs[r*LS + c0    ], gp);
        async_g2l_b128(&xs[r*LS + c0 + 4], gp + 4);
    }
    wait_async0();
    __syncthreads();

    // ---- lin = x@M.T via fp32 WMMA ----
    v8f acc = {};
    const float* Mp = M + (size_t)(wv*16 + lr) * DIM + 2*hh;
    const float* Ap = &xs[lr*LS + 2*hh];
    __builtin_prefetch(Mp, 0, 0);
#pragma unroll 8
    for (int kt = 0; kt < 32; ++kt) {
        v2f a  = *(const v2f*)(Ap + kt*4);
        v2f bm = *(const v2f*)(Mp + kt*4);
        acc = __builtin_amdgcn_wmma_f32_16x16x4_f32(false, a, false, bm,
                                                    (short)0, acc, false, false);
    }
#pragma unroll
    for (int v = 0; v < 8; ++v) lin[(v + 8*hh)*LS + wv*16 + lr] = acc[v];
    __syncthreads();

    // ---- layernorm -> xt ----
    const int row = tid >> 4;
    const int cl  = tid & 15;
    float vals[8];
    float s1 = 0.f, s2 = 0.f;
#pragma unroll
    for (int q = 0; q < 8; ++q) {
        float vL = lin[row*LS + cl + q*16];
        vals[q] = vL; s1 += vL; s2 += vL*vL;
    }
#pragma unroll
    for (int off = 8; off; off >>= 1) {
        s1 += __shfl_xor(s1, off, 16);
        s2 += __shfl_xor(s2, off, 16);
    }
    const float mu   = s1 * (1.f/DIM);
    const float rstd = rsqrtf(s2*(1.f/DIM) - mu*mu + 1e-5f);
#pragma unroll
    for (int q = 0; q < 8; ++q) {
        const int cc = cl + q*16;
        xt[row*LS + cc] = (vals[q] - mu) * rstd * g[cc] + b[cc];
    }
    __syncthreads();

    // ---- Nk in 8 i-chunks of 16; Peff shared across the 8 batch rows ----
    const int pr  = tid >> 3;          // 0..31 (ss, i_local) producer pair
    const int ssP = pr >> 4;
    const int ilP = pr & 15;
    const int jb  = (tid & 7) * 16;    // 16 j's per producer thread
    const int mO  = tid >> 4;          // consumer: output row
    const int iO  = tid & 15;          // consumer: output i within chunk
    const int ssO = mO >> 3;
    const size_t tokO = (size_t)(mO & 7) * SEQ + (size_t)(s0 + ssO);

    for (int ic = 0; ic < 8; ++ic) {
        const int   i    = ic*16 + ilP;
        const float sv   = (float)(s0 + ssP);
        const float* Prw = P + (size_t)i * DIM + jb;
        float*       pe  = &peff[ssP*2080 + ilP*130 + jb];
#pragma unroll
        for (int jj = 0; jj < 16; ++jj) {
            const float period = (float)(i*DIM + jb + jj + 2);
            pe[jj] = Prw[jj] * cos2pi(sv / period);
        }
        __syncthreads();

        float a2 = 0.f;
        const float* xr = &xt[mO*LS];
        const float* pp = &peff[ssO*2080 + iO*130];
#pragma unroll 8
        for (int jj2 = 0; jj2 < DIM; ++jj2) a2 += xr[jj2] * pp[jj2];

        const int iG = ic*16 + iO;
        out[tokO * DIM + iG] = a2 + lin[mO*LS + iG];
        __syncthreads();   // protect peff before next chunk overwrites it
    }
}

// ---------------------------------------------------------------------------
extern "C" void kernel_launch(void* const* d_in, const int* in_sizes, int n_in,
                              void* d_out, int out_size, void* d_ws, size_t ws_size,
                              hipStream_t stream) {
    const float* x   = (const float*)d_in[0];
    const float* W0  = (const float*)d_in[1];
    const float* R0  = (const float*)d_in[2];
    const float* A0  = (const float*)d_in[3];
    const float* g0  = (const float*)d_in[4];
    const float* b0  = (const float*)d_in[5];
    const float* M1  = (const float*)d_in[6];
    const float* P1  = (const float*)d_in[7];
    const float* g1  = (const float*)d_in[8];
    const float* b1  = (const float*)d_in[9];
    const float* W2  = (const float*)d_in[10];
    const float* R2  = (const float*)d_in[11];
    const float* A2  = (const float*)d_in[12];
    const float* g2  = (const float*)d_in[13];
    const float* b2  = (const float*)d_in[14];
    const float* M3  = (const float*)d_in[15];
    const float* P3  = (const float*)d_in[16];
    const float* g3  = (const float*)d_in[17];
    const float* b3  = (const float*)d_in[18];

    float* out = (float*)d_out;
    float* h   = (float*)d_ws;   // one (8,4096,128) fp32 intermediate = 16 MB

    const dim3 blk(256);
    const int TOK_TILES = (BATCH * SEQ) / 16;  // 2048
    const int PM_TILES  = SEQ / 2;             // 2048

    // x -> ws -> out -> ws -> out
    k_ab<<<TOK_TILES, blk, 0, stream>>>(x,   W0, R0, A0, g0, b0, h);
    k_pm<<<PM_TILES,  blk, 0, stream>>>(h,   M1, P1, g1, b1, out);
    k_ab<<<TOK_TILES, blk, 0, stream>>>(out, W2, R2, A2, g2, b2, h);
    k_pm<<<PM_TILES,  blk, 0, stream>>>(h,   M3, P3, g3, b3, out);
}